// SlotAttention_62199716381058
// MI455X (gfx1250) — compile-verified
//
#include <hip/hip_runtime.h>

#define B_ 64
#define N_ 4096
#define F_ 256
#define D_ 256
#define S_ 8
#define H_ 512
#define EPS_ 1e-8f

typedef float v2f __attribute__((ext_vector_type(2)));
typedef float v8f __attribute__((ext_vector_type(8)));

static __device__ __forceinline__ v8f wmma4(v2f a, v2f b, v8f c) {
    // D = A(16x4 f32) x B(4x16 f32) + C(16x16 f32)
    return __builtin_amdgcn_wmma_f32_16x16x4_f32(false, a, false, b, (short)0, c, false, false);
}

// ---------------------------------------------------------------------------
// slots = mu + exp(log_sigma) * noise
__global__ void k_init_slots(const float* __restrict__ noise, const float* __restrict__ mu,
                             const float* __restrict__ ls, float* __restrict__ slots) {
    int i = blockIdx.x * 256 + threadIdx.x;  // B*S*D = 131072
    int d = i & (D_ - 1);
    slots[i] = mu[d] + expf(ls[d]) * noise[i];
}

// ---------------------------------------------------------------------------
// fused LayerNorm(x) + k = xn@Wk^T, v = xn@Wv^T   (f32 WMMA)
// grid: B*N/128 blocks, 256 threads (8 waves); each wave owns 16 rows.
__global__ __launch_bounds__(256) void k_ln_kv(const float* __restrict__ x,
                                               const float* __restrict__ g,
                                               const float* __restrict__ bta,
                                               const float* __restrict__ Wk,
                                               const float* __restrict__ Wv,
                                               float* __restrict__ kout,
                                               float* __restrict__ vout) {
    __shared__ float xn[128 * F_];  // 128 KB LDS tile
    const int wave = threadIdx.x >> 5, lane = threadIdx.x & 31;
    const long row0 = (long)blockIdx.x * 128;

    // --- LayerNorm: wave handles rows [wave*16, wave*16+16)
    for (int rr = 0; rr < 16; ++rr) {
        int r = wave * 16 + rr;
        const float* xr = x + (row0 + r) * F_;
        float vals[8];
        float s = 0.f, s2 = 0.f;
#pragma unroll
        for (int i = 0; i < 8; ++i) {
            float t = xr[lane + 32 * i];
            vals[i] = t; s += t; s2 += t * t;
        }
#pragma unroll
        for (int off = 16; off >= 1; off >>= 1) {
            s += __shfl_xor(s, off);
            s2 += __shfl_xor(s2, off);
        }
        float mean = s * (1.0f / F_);
        float var  = s2 * (1.0f / F_) - mean * mean;
        float rstd = rsqrtf(var + 1e-5f);
#pragma unroll
        for (int i = 0; i < 8; ++i) {
            int c = lane + 32 * i;
            xn[r * F_ + c] = (vals[i] - mean) * rstd * g[c] + bta[c];
        }
    }
    __syncthreads();

    // --- GEMM: wave computes its 16-row M-tile x 256 cols, for Wk then Wv
    const int m = lane & 15, h = lane >> 4;
    const float* xrow = xn + (wave * 16 + m) * F_;
    const v8f vzero = {0.f, 0.f, 0.f, 0.f, 0.f, 0.f, 0.f, 0.f};
    for (int tgt = 0; tgt < 2; ++tgt) {
        const float* W = tgt ? Wv : Wk;
        float* out = tgt ? vout : kout;
        v8f acc[16];
#pragma unroll
        for (int nt = 0; nt < 16; ++nt) acc[nt] = vzero;
#pragma unroll 1
        for (int kb = 0; kb < F_; kb += 4) {
            v2f a = *(const v2f*)(xrow + kb + 2 * h);   // A[m][kb+2h], A[m][kb+2h+1]
#pragma unroll
            for (int nt = 0; nt < 16; ++nt) {
                v2f bf = *(const v2f*)(W + (nt * 16 + m) * F_ + kb + 2 * h);  // B[k][n]=W[n][f]
                acc[nt] = wmma4(a, bf, acc[nt]);
            }
        }
#pragma unroll
        for (int nt = 0; nt < 16; ++nt) {
#pragma unroll
            for (int e = 0; e < 8; ++e) {
                int mr = e + 8 * h;
                out[(row0 + wave * 16 + mr) * D_ + nt * 16 + m] = acc[nt][e];
            }
        }
    }
}

// ---------------------------------------------------------------------------
__global__ void k_zero(float* __restrict__ p, int n) {
    int i = blockIdx.x * 256 + threadIdx.x;
    if (i < n) p[i] = 0.f;
}

// ---------------------------------------------------------------------------
// s = LN(slots over S*D) ; q = (s @ Wq^T) * D^-0.5     (one block per batch)
__global__ __launch_bounds__(256) void k_slots_q(const float* __restrict__ slots,
                                                 const float* __restrict__ g,
                                                 const float* __restrict__ bt,
                                                 const float* __restrict__ Wq,
                                                 float* __restrict__ q) {
    __shared__ float sl[S_ * D_];
    __shared__ float red[32];
    int b = blockIdx.x, t = threadIdx.x;
    const float* sb = slots + b * (S_ * D_);
    float s = 0.f, s2 = 0.f;
#pragma unroll
    for (int j = 0; j < 8; ++j) {
        float v = sb[t + 256 * j];
        sl[t + 256 * j] = v; s += v; s2 += v * v;
    }
    for (int off = 16; off >= 1; off >>= 1) { s += __shfl_xor(s, off); s2 += __shfl_xor(s2, off); }
    int wave = t >> 5, lane = t & 31;
    if (lane == 0) { red[wave] = s; red[8 + wave] = s2; }
    __syncthreads();
    if (t == 0) {
        float a = 0.f, c = 0.f;
        for (int w = 0; w < 8; ++w) { a += red[w]; c += red[8 + w]; }
        red[16] = a; red[17] = c;
    }
    __syncthreads();
    float mean = red[16] * (1.0f / (S_ * D_));
    float var  = red[17] * (1.0f / (S_ * D_)) - mean * mean;
    float rstd = rsqrtf(var + 1e-5f);
#pragma unroll
    for (int j = 0; j < 8; ++j) {
        int idx = t + 256 * j;
        sl[idx] = (sl[idx] - mean) * rstd * g[idx] + bt[idx];
    }
    __syncthreads();
    const float scale = 0.0625f;  // 256^-0.5
#pragma unroll 1
    for (int j = 0; j < 8; ++j) {
        int idx = t + 256 * j;
        int ss = idx >> 8, d = idx & 255;
        const float* wrow = Wq + d * D_;
        const float* srow = sl + ss * D_;
        float acc = 0.f;
        for (int dd = 0; dd < D_; ++dd) acc += srow[dd] * wrow[dd];
        q[b * (S_ * D_) + idx] = acc * scale;
    }
}

// ---------------------------------------------------------------------------
// logits = k @ q^T (S padded to 16), softmax over S, +EPS, partial colsums.
// grid: B*N/128 blocks of 256 threads; each wave owns a 16-row tile.
__global__ __launch_bounds__(256) void k_logits(const float* __restrict__ kbuf,
                                                const float* __restrict__ q,
                                                float* __restrict__ attn,
                                                float* __restrict__ colsum) {
    __shared__ float qs[S_ * D_];
    int t = threadIdx.x;
    long r0 = (long)blockIdx.x * 128;  // flat row over B*N
    int b = (int)(r0 >> 12);           // N = 4096
#pragma unroll
    for (int j = 0; j < 8; ++j) qs[t + 256 * j] = q[b * (S_ * D_) + t + 256 * j];
    __syncthreads();

    int wave = t >> 5, lane = t & 31, m = lane & 15, h = lane >> 4;
    const float* krow = kbuf + (r0 + wave * 16 + m) * D_;
    v8f acc = {0.f, 0.f, 0.f, 0.f, 0.f, 0.f, 0.f, 0.f};
    bool valid = (m < S_);
#pragma unroll 4
    for (int kb = 0; kb < D_; kb += 4) {
        v2f a = *(const v2f*)(krow + kb + 2 * h);
        v2f bf;
        if (valid) bf = *(const v2f*)(qs + m * D_ + kb + 2 * h);
        else { bf.x = 0.f; bf.y = 0.f; }
        acc = wmma4(a, bf, acc);
    }

    float csum = 0.f;
#pragma unroll
    for (int e = 0; e < 8; ++e) {
        float val = acc[e];
        float mx = val;
        mx = fmaxf(mx, __shfl_xor(mx, 1));
        mx = fmaxf(mx, __shfl_xor(mx, 2));
        mx = fmaxf(mx, __shfl_xor(mx, 4));
        float ex = __expf(val - mx);
        float sm = ex;
        sm += __shfl_xor(sm, 1); sm += __shfl_xor(sm, 2); sm += __shfl_xor(sm, 4);
        float at = ex / sm + EPS_;
        int row = wave * 16 + e + 8 * h;
        if (m < S_) attn[(r0 + row) * S_ + m] = at;
        csum += (m < S_) ? at : 0.f;
    }
    csum += __shfl_xor(csum, 16);  // combine the two row-halves (same s)
    if (lane < S_) atomicAdd(&colsum[b * S_ + lane], csum);
}

// ---------------------------------------------------------------------------
// updates_raw = attn^T @ v   (S padded to 16); grid: (16 d-tiles, B batches)
__global__ __launch_bounds__(256) void k_updates(const float* __restrict__ attn,
                                                 const float* __restrict__ vbuf,
                                                 float* __restrict__ upd) {
    __shared__ float red[8 * 256];
    int b = blockIdx.y;
    int d0 = blockIdx.x * 16;
    int t = threadIdx.x, wave = t >> 5, lane = t & 31, m = lane & 15, h = lane >> 4;
    const float* attnB = attn + (long)b * N_ * S_;
    const float* vB = vbuf + (long)b * N_ * D_;
    v8f acc = {0.f, 0.f, 0.f, 0.f, 0.f, 0.f, 0.f, 0.f};
    int n0 = wave * 512;
#pragma unroll 2
    for (int nb = n0; nb < n0 + 512; nb += 4) {
        v2f a, bf;
        if (m < S_) {
            a.x = attnB[(nb + 2 * h) * S_ + m];      // A[m][k] = attn[n][s]
            a.y = attnB[(nb + 2 * h + 1) * S_ + m];
        } else { a.x = 0.f; a.y = 0.f; }
        const float* vp = vB + (long)(nb + 2 * h) * D_ + d0 + m;  // B[k][j] = v[n][d0+j]
        bf.x = vp[0];
        bf.y = vp[D_];
        acc = wmma4(a, bf, acc);
    }
#pragma unroll
    for (int e = 0; e < 8; ++e) red[wave * 256 + e * 32 + lane] = acc[e];
    __syncthreads();
    float ssum = 0.f;
#pragma unroll
    for (int w = 0; w < 8; ++w) ssum += red[w * 256 + t];
    int e = t >> 5, ln = t & 31;
    int mm = e + 8 * (ln >> 4), dd = ln & 15;
    if (mm < S_) upd[b * (S_ * D_) + mm * D_ + d0 + dd] = ssum;
}

// ---------------------------------------------------------------------------
// GRU cell + LN(m over S*D) + MLP residual.  One block per batch.
__global__ __launch_bounds__(256) void k_gru_mlp(
    const float* __restrict__ upd, const float* __restrict__ colsum,
    const float* __restrict__ slots_in,
    const float* __restrict__ W_ih, const float* __restrict__ W_hh,
    const float* __restrict__ b_ih, const float* __restrict__ b_hh,
    const float* __restrict__ lng, const float* __restrict__ lnb,
    const float* __restrict__ W1, const float* __restrict__ b1,
    const float* __restrict__ W2, const float* __restrict__ b2,
    float* __restrict__ slots_out, float* __restrict__ final_out) {
    __shared__ float u[2048];   // updates (renormalized), later reused as m
    __shared__ float hp[2048];  // h_prev
    __shared__ float sn[2048];  // new slots
    __shared__ float hm[4096];  // mlp hidden
    __shared__ float red[32];
    int b = blockIdx.x, t = threadIdx.x;
#pragma unroll
    for (int j = 0; j < 8; ++j) {
        int idx = t + 256 * j;
        int ss = idx >> 8;
        hp[idx] = slots_in[b * 2048 + idx];
        u[idx] = upd[b * 2048 + idx] / colsum[b * S_ + ss];  // deferred renormalization
    }
    __syncthreads();
    // --- GRU (gate order r,z,n)
#pragma unroll 1
    for (int j = 0; j < 8; ++j) {
        int idx = t + 256 * j;
        int ss = idx >> 8, d = idx & 255;
        const float* ur = u + ss * 256;
        const float* hr = hp + ss * 256;
        float ir = b_ih[d], iz = b_ih[256 + d], in_ = b_ih[512 + d];
        float gr = b_hh[d], gz = b_hh[256 + d], gn = b_hh[512 + d];
        const float* wir = W_ih + d * 256;
        const float* wiz = W_ih + (256 + d) * 256;
        const float* win = W_ih + (512 + d) * 256;
        const float* whr = W_hh + d * 256;
        const float* whz = W_hh + (256 + d) * 256;
        const float* whn = W_hh + (512 + d) * 256;
        for (int dd = 0; dd < 256; ++dd) {
            float uu = ur[dd], hh = hr[dd];
            ir += uu * wir[dd]; iz += uu * wiz[dd]; in_ += uu * win[dd];
            gr += hh * whr[dd]; gz += hh * whz[dd]; gn += hh * whn[dd];
        }
        float r = 1.f / (1.f + __expf(-(ir + gr)));
        float z = 1.f / (1.f + __expf(-(iz + gz)));
        float nn = tanhf(in_ + r * gn);
        sn[idx] = (1.f - z) * nn + z * hr[d];
    }
    __syncthreads();
    // --- LayerNorm over S*D
    float s = 0.f, s2 = 0.f;
#pragma unroll
    for (int j = 0; j < 8; ++j) { float v = sn[t + 256 * j]; s += v; s2 += v * v; }
    for (int off = 16; off >= 1; off >>= 1) { s += __shfl_xor(s, off); s2 += __shfl_xor(s2, off); }
    int wave = t >> 5, lane = t & 31;
    if (lane == 0) { red[wave] = s; red[8 + wave] = s2; }
    __syncthreads();
    if (t == 0) {
        float a = 0.f, c = 0.f;
        for (int w = 0; w < 8; ++w) { a += red[w]; c += red[8 + w]; }
        red[16] = a; red[17] = c;
    }
    __syncthreads();
    float mean = red[16] * (1.f / 2048.f);
    float var  = red[17] * (1.f / 2048.f) - mean * mean;
    float rstd = rsqrtf(var + 1e-5f);
#pragma unroll
    for (int j = 0; j < 8; ++j) {
        int idx = t + 256 * j;
        u[idx] = (sn[idx] - mean) * rstd * lng[idx] + lnb[idx];  // m
    }
    __syncthreads();
    // --- MLP layer 1: relu(m @ W1^T + b1)
#pragma unroll 1
    for (int j = 0; j < 16; ++j) {
        int idx = t + 256 * j;
        int ss = idx >> 9, hc = idx & 511;
        const float* mr = u + ss * 256;
        const float* w = W1 + hc * 256;
        float a = b1[hc];
        for (int dd = 0; dd < 256; ++dd) a += mr[dd] * w[dd];
        hm[idx] = fmaxf(a, 0.f);
    }
    __syncthreads();
    // --- MLP layer 2 + residual
#pragma unroll 1
    for (int j = 0; j < 8; ++j) {
        int idx = t + 256 * j;
        int ss = idx >> 8, d = idx & 255;
        const float* hr2 = hm + ss * 512;
        const float* w = W2 + d * 512;
        float a = b2[d];
        for (int dd = 0; dd < 512; ++dd) a += hr2[dd] * w[dd];
        float outv = sn[idx] + a;
        slots_out[b * 2048 + idx] = outv;
        if (final_out) final_out[b * 2048 + idx] = outv;
    }
}

// ---------------------------------------------------------------------------
extern "C" void kernel_launch(void* const* d_in, const int* in_sizes, int n_in,
                              void* d_out, int out_size, void* d_ws, size_t ws_size,
                              hipStream_t stream) {
    const float* x        = (const float*)d_in[0];
    const float* noise    = (const float*)d_in[1];
    const float* ln_in_g  = (const float*)d_in[2];
    const float* ln_in_b  = (const float*)d_in[3];
    const float* ln_s_g   = (const float*)d_in[4];
    const float* ln_s_b   = (const float*)d_in[5];
    const float* ln_m_g   = (const float*)d_in[6];
    const float* ln_m_b   = (const float*)d_in[7];
    const float* slots_mu = (const float*)d_in[8];
    const float* slots_ls = (const float*)d_in[9];
    const float* Wq       = (const float*)d_in[10];
    const float* Wk       = (const float*)d_in[11];
    const float* Wv       = (const float*)d_in[12];
    const float* W_ih     = (const float*)d_in[13];
    const float* W_hh     = (const float*)d_in[14];
    const float* b_ih     = (const float*)d_in[15];
    const float* b_hh     = (const float*)d_in[16];
    const float* W1       = (const float*)d_in[17];
    const float* b1       = (const float*)d_in[18];
    const float* W2       = (const float*)d_in[19];
    const float* b2       = (const float*)d_in[20];
    float* out = (float*)d_out;

    float* ws = (float*)d_ws;
    size_t o = 0;
    float* kbuf = ws + o;   o += (size_t)B_ * N_ * D_;   // 67.1M
    float* vbuf = ws + o;   o += (size_t)B_ * N_ * D_;   // 67.1M
    float* attn = ws + o;   o += (size_t)B_ * N_ * S_;   // 2.1M
    float* q    = ws + o;   o += (size_t)B_ * S_ * D_;
    float* slots= ws + o;   o += (size_t)B_ * S_ * D_;
    float* upd  = ws + o;   o += (size_t)B_ * S_ * D_;
    float* csum = ws + o;   o += (size_t)B_ * S_;

    k_init_slots<<<(B_ * S_ * D_) / 256, 256, 0, stream>>>(noise, slots_mu, slots_ls, slots);
    k_ln_kv<<<(B_ * N_) / 128, 256, 0, stream>>>(x, ln_in_g, ln_in_b, Wk, Wv, kbuf, vbuf);

    for (int it = 0; it < 3; ++it) {
        k_slots_q<<<B_, 256, 0, stream>>>(slots, ln_s_g, ln_s_b, Wq, q);
        k_zero<<<2, 256, 0, stream>>>(csum, B_ * S_);
        k_logits<<<(B_ * N_) / 128, 256, 0, stream>>>(kbuf, q, attn, csum);
        k_updates<<<dim3(16, B_), 256, 0, stream>>>(attn, vbuf, upd);
        k_gru_mlp<<<B_, 256, 0, stream>>>(upd, csum, slots, W_ih, W_hh, b_ih, b_hh,
                                          ln_m_g, ln_m_b, W1, b1, W2, b2,
                                          slots, (it == 2) ? out : nullptr);
    }
}